// ContextualContinuityLoss_33543694581774
// MI455X (gfx1250) — compile-verified
//
#include <hip/hip_runtime.h>
#include <stdint.h>

// ContextualContinuityLoss on MI455X (gfx1250):
//   loss = sum over (b, s) of ||emb[b,s]-emb[b,s+1]|| where labels match & != 0,
//   divided by B*S.  B=32, S=4096, D=256 (fp32).
// Pure streaming-bandwidth problem (128 MB read, ~0.2 GFLOP -> 23.3 TB/s floor
// ~5.5us; WMMA has no roofline justification). Strategy:
//   - each wave streams 33 rows (32 pairs) through a 4-deep async LDS pipeline
//     using GLOBAL_LOAD_ASYNC_TO_LDS_B128 + s_wait_asynccnt (CDNA5 async path);
//     4 KB in flight per wave x 4096 waves = 16 MB outstanding ~ HBM BDP
//   - previous row kept in registers -> every row read from HBM exactly once
//   - wave32 ballot for the label mask, shfl_xor for the 32-lane dot reduction
//   - deterministic two-pass reduction (per-wave partials in d_ws, fixed-order sum)

#define B_DIM 32
#define S_DIM 4096
#define D_DIM 256
#define ROW_BYTES 1024           // 256 * 4
#define PAIRS_PER_WAVE 32
#define CHUNKS_PER_ROW 128       // ceil(4095 / 32)
#define TOTAL_WAVES (B_DIM * CHUNKS_PER_ROW)   // 4096
#define WAVES_PER_BLOCK 8
#define NBUF 4                   // power of two -> buffer index is an AND

// --- CDNA5 async global->LDS helpers (inline asm; no builtin on ROCm 7.2) ---

// Issue one full 1KB row: 32 lanes x 16B x 2 issues. ldsAddr/g are per-lane.
__device__ __forceinline__ void async_row(const float* g, unsigned ldsAddr) {
    asm volatile("global_load_async_to_lds_b128 %0, %1, off"
                 :: "v"(ldsAddr), "v"(g) : "memory");
    asm volatile("global_load_async_to_lds_b128 %0, %1, off"
                 :: "v"(ldsAddr + 512u), "v"(g + 128) : "memory");
}

__device__ __forceinline__ void wait_async_le6() {
    asm volatile("s_wait_asynccnt 0x6" ::: "memory");
}
__device__ __forceinline__ void wait_async_le4() {
    asm volatile("s_wait_asynccnt 0x4" ::: "memory");
}
__device__ __forceinline__ void wait_async_le2() {
    asm volatile("s_wait_asynccnt 0x2" ::: "memory");
}
__device__ __forceinline__ void wait_async_le0() {
    asm volatile("s_wait_asynccnt 0x0" ::: "memory");
}

__global__ __launch_bounds__(WAVES_PER_BLOCK * 32)
void ccl_partial_kernel(const float* __restrict__ emb,
                        const long long* __restrict__ lab,
                        float* __restrict__ partials)
{
    __shared__ char smem[WAVES_PER_BLOCK * NBUF * ROW_BYTES];   // 32 KB

    const int lane      = threadIdx.x & 31;
    const int waveInBlk = threadIdx.x >> 5;
    const int w  = blockIdx.x * WAVES_PER_BLOCK + waveInBlk;   // 0..4095
    const int b  = w >> 7;                                     // batch row
    const int c  = w & (CHUNKS_PER_ROW - 1);
    const int p0 = c * PAIRS_PER_WAVE;                         // first pair idx
    const int rem = (S_DIM - 1) - p0;
    const int cnt = rem < PAIRS_PER_WAVE ? rem : PAIRS_PER_WAVE; // 31 or 32

    // ---- label mask for this chunk: one ballot, reused for all 32 pairs ----
    unsigned mask;
    {
        bool bit = false;
        if (lane < cnt) {
            const long long* L = lab + ((size_t)b << 12) + p0;
            long long l0 = L[lane];
            long long l1 = L[lane + 1];
            bit = (l0 != 0) && (l0 == l1);
        }
        mask = (unsigned)__ballot(bit);   // wave32: low 32 bits
    }

    // ---- LDS byte offsets for this wave's 4 private row buffers ----
    char* base = &smem[waveInBlk * NBUF * ROW_BYTES];
    const unsigned ldsBase = (unsigned)(uintptr_t)base;  // low 32 bits = LDS offset
    const unsigned laneLds = ldsBase + (unsigned)(lane * 16);

    // per-lane global pointer into row 0 of this chunk
    const float* rowBase = emb + (((size_t)b << 12) + (size_t)p0) * D_DIM;
    const float* gLane   = rowBase + lane * 4;

    // ---- prologue: prefetch rows 0..3 (cnt >= 31 so always valid) ----
    async_row(gLane + 0 * D_DIM, laneLds + 0 * ROW_BYTES);
    async_row(gLane + 1 * D_DIM, laneLds + 1 * ROW_BYTES);
    async_row(gLane + 2 * D_DIM, laneLds + 2 * ROW_BYTES);
    async_row(gLane + 3 * D_DIM, laneLds + 3 * ROW_BYTES);

    wait_async_le6();   // row 0 complete (2 ops), rows 1..3 still in flight

    // lane's 8 floats (32B) of the "previous" row, kept in registers
    const float4* rp = (const float4*)(base + 0 * ROW_BYTES + lane * 32);
    float4 pA = rp[0];
    float4 pB = rp[1];

    float wsum = 0.0f;

    for (int i = 0; i < cnt; ++i) {
        const bool pf4 = (i + 4 <= cnt);
        const bool pf3 = (i + 3 <= cnt);
        const bool pf2 = (i + 2 <= cnt);

        if (pf4) {
            // buf[i&3] held row i, already consumed into registers last iter
            async_row(gLane + (size_t)(i + 4) * D_DIM,
                      laneLds + (unsigned)((i & 3) * ROW_BYTES));
        }
        // gate on row i+1 being resident in LDS (2 async ops per row)
        if (pf4)      wait_async_le6();   // in flight: rows i+1..i+4 (8 ops)
        else if (pf3) wait_async_le4();   // in flight: rows i+1..i+3 (6 ops)
        else if (pf2) wait_async_le2();   // in flight: rows i+1..i+2 (4 ops)
        else          wait_async_le0();   // in flight: row i+1 only -> drain

        const float4* cp =
            (const float4*)(base + ((i + 1) & 3) * ROW_BYTES + lane * 32);
        float4 cA = cp[0];
        float4 cB = cp[1];

        float dx, acc;
        dx = cA.x - pA.x; acc  = dx * dx;
        dx = cA.y - pA.y; acc += dx * dx;
        dx = cA.z - pA.z; acc += dx * dx;
        dx = cA.w - pA.w; acc += dx * dx;
        dx = cB.x - pB.x; acc += dx * dx;
        dx = cB.y - pB.y; acc += dx * dx;
        dx = cB.z - pB.z; acc += dx * dx;
        dx = cB.w - pB.w; acc += dx * dx;

        // wave32 cross-lane reduction (5 steps)
        acc += __shfl_xor(acc, 16, 32);
        acc += __shfl_xor(acc,  8, 32);
        acc += __shfl_xor(acc,  4, 32);
        acc += __shfl_xor(acc,  2, 32);
        acc += __shfl_xor(acc,  1, 32);

        if ((mask >> i) & 1u)
            wsum += sqrtf(acc);

        pA = cA;
        pB = cB;
    }

    if (lane == 0)
        partials[w] = wsum;
}

// Deterministic fixed-order reduction of the 4096 wave partials.
__global__ __launch_bounds__(256)
void ccl_reduce_kernel(const float* __restrict__ partials,
                       float* __restrict__ out)
{
    __shared__ float red[256];
    const int t = threadIdx.x;
    float s = 0.0f;
    for (int k = t; k < TOTAL_WAVES; k += 256)
        s += partials[k];
    red[t] = s;
    __syncthreads();
    for (int off = 128; off > 0; off >>= 1) {
        if (t < off) red[t] += red[t + off];
        __syncthreads();
    }
    if (t == 0)
        out[0] = red[0] * (1.0f / (float)(B_DIM * S_DIM));
}

extern "C" void kernel_launch(void* const* d_in, const int* in_sizes, int n_in,
                              void* d_out, int out_size, void* d_ws, size_t ws_size,
                              hipStream_t stream) {
    (void)in_sizes; (void)n_in; (void)out_size; (void)ws_size;

    const float*     emb = (const float*)d_in[0];        // (32, 4096, 256) fp32
    const long long* lab = (const long long*)d_in[1];    // (32, 4096) int64
    float* out      = (float*)d_out;                     // scalar fp32
    float* partials = (float*)d_ws;                      // 4096 floats scratch

    const int blocks = TOTAL_WAVES / WAVES_PER_BLOCK;    // 512
    ccl_partial_kernel<<<blocks, WAVES_PER_BLOCK * 32, 0, stream>>>(emb, lab, partials);
    ccl_reduce_kernel<<<1, 256, 0, stream>>>(partials, out);
}